// SimpleMPNN_37168646980027
// MI455X (gfx1250) — compile-verified
//
#include <hip/hip_runtime.h>

typedef float v2f __attribute__((ext_vector_type(2)));
typedef float v8f __attribute__((ext_vector_type(8)));

#define DIM   128
#define ALPHA 0.5f
#define EPS   1e-12f

// ---------------- zero fill (grid-stride) ----------------
__global__ void k_zero(float* __restrict__ p, long n) {
  long i = (long)blockIdx.x * blockDim.x + threadIdx.x;
  long stride = (long)gridDim.x * blockDim.x;
  for (; i < n; i += stride) p[i] = 0.0f;
}

// ---------------- normalize question embedding (1 block, 128 thr) ----------------
__global__ void k_qnorm(const float* __restrict__ q, float* __restrict__ qn) {
  __shared__ float red[DIM];
  int t = threadIdx.x;
  float v = q[t];
  red[t] = v * v;
  __syncthreads();
  for (int s = DIM / 2; s > 0; s >>= 1) {
    if (t < s) red[t] += red[t + s];
    __syncthreads();
  }
  float n = sqrtf(red[0]);
  qn[t] = v / fmaxf(n, EPS);
}

// ---------------- in-degree via f32 atomics ----------------
__global__ void k_deg(const long long* __restrict__ dst, float* __restrict__ deg, int E) {
  int e = blockIdx.x * blockDim.x + threadIdx.x;
  if (e < E) unsafeAtomicAdd(&deg[(int)dst[e]], 1.0f);
}

// ---------------- edge similarity via V_WMMA_F32_16X16X4_F32 ----------------
// One wave handles 16 edges. A = 16 rows of edge_attr, B = q broadcast into all
// 16 columns; chained accumulation over K = 128 in chunks of 4.
// D[m][n] = row_m . q for all n.  Row norms accumulated in VALU, halves combined
// with a cross-half shuffle (wave32).
__global__ void __launch_bounds__(256)
k_edge_sim(const float* __restrict__ edge_attr,
           const float* __restrict__ qn,
           float* __restrict__ sim, int E) {
  __shared__ float s_q[DIM];
  __shared__ float s_dot[8][16];

  int tid = threadIdx.x;
  if (tid < DIM) s_q[tid] = qn[tid];
  __syncthreads();

  int waveInBlk = tid >> 5;
  int lane = tid & 31;
  int h = lane >> 4;   // half of the wave -> K sub-slot
  int j = lane & 15;   // row (edge) within the 16-edge tile

  long gwave = (long)blockIdx.x * 8 + waveInBlk;
  long base  = gwave * 16;

  long row = base + j;
  if (row >= E) row = (long)E - 1;           // clamp: keep EXEC all-ones for WMMA
  const float* rp = edge_attr + row * (long)DIM;

  v8f acc = {0.f, 0.f, 0.f, 0.f, 0.f, 0.f, 0.f, 0.f};
  float nrm = 0.0f;

#pragma unroll
  for (int c = 0; c < DIM / 4; ++c) {
    int k = 4 * c + 2 * h;                    // ISA A layout: slot(v,h) = K = 2h+v
    v2f a = { rp[k],  rp[k + 1] };
    v2f b = { s_q[k], s_q[k + 1] };           // same K slot-mapping for B
    nrm = fmaf(a.x, a.x, nrm);
    nrm = fmaf(a.y, a.y, nrm);
    acc = __builtin_amdgcn_wmma_f32_16x16x4_f32(
        /*neg_a=*/false, a, /*neg_b=*/false, b,
        /*c_mod=*/(short)0, acc, /*reuse_a=*/false, /*reuse_b=*/false);
  }

  // full ||row||^2 for row (lane&15) in every lane
  nrm += __shfl_xor(nrm, 16, 32);

  // D layout: VGPR v, lanes 0-15 -> M=v ; lanes 16-31 -> M=v+8 (all N identical)
  if (j == 0) {
#pragma unroll
    for (int v = 0; v < 8; ++v) s_dot[waveInBlk][v + 8 * h] = acc[v];
  }
  __syncthreads();

  long e = base + j;
  if (h == 0 && e < E) {
    float dot = s_dot[waveInBlk][j];
    sim[e] = dot / fmaxf(sqrtf(nrm), EPS);
  }
}

// ---------------- gather + scale + scatter-add (one wave per edge) ----------------
__global__ void k_scatter(const float* __restrict__ x,
                          const long long* __restrict__ src,
                          const long long* __restrict__ dst,
                          const float* __restrict__ sim,
                          float* __restrict__ xnew, int E) {
  long tid = (long)blockIdx.x * blockDim.x + threadIdx.x;
  int e = (int)(tid >> 5);
  if (e >= E) return;
  int lane = (int)(tid & 31);
  int c = lane * 4;                            // 32 lanes x float4 = 128 floats
  long s = src[e];
  long d = dst[e];
  float w = sim[e];
  float4 v = *(const float4*)(x + s * (long)DIM + c);
  float* out = xnew + d * (long)DIM + c;
  unsafeAtomicAdd(out + 0, v.x * w);
  unsafeAtomicAdd(out + 1, v.y * w);
  unsafeAtomicAdd(out + 2, v.z * w);
  unsafeAtomicAdd(out + 3, v.w * w);
}

// ---------------- x_out = a*x + (1-a)*x_new/deg ----------------
__global__ void k_update(const float* __restrict__ xcur,
                         const float* __restrict__ xnew,
                         const float* __restrict__ deg,
                         float* __restrict__ xout, long n) {
  long i = (long)blockIdx.x * blockDim.x + threadIdx.x;
  if (i >= n) return;
  float dg = fmaxf(deg[i >> 7], 1.0f);
  xout[i] = ALPHA * xcur[i] + (1.0f - ALPHA) * (xnew[i] / dg);
}

extern "C" void kernel_launch(void* const* d_in, const int* in_sizes, int n_in,
                              void* d_out, int out_size, void* d_ws, size_t ws_size,
                              hipStream_t stream) {
  const float*     x          = (const float*)d_in[0];
  const long long* edge_index = (const long long*)d_in[1];   // int64 per reference
  const float*     edge_attr  = (const float*)d_in[2];
  const float*     q          = (const float*)d_in[3];

  const int N = in_sizes[0] / DIM;     // 50000
  const int E = in_sizes[1] / 2;       // 600000
  const long long* src = edge_index;       // row 0
  const long long* dst = edge_index + E;   // row 1

  float* ws   = (float*)d_ws;
  float* qn   = ws;                        // 128
  float* deg  = ws + DIM;                  // N
  float* sim  = deg + N;                   // E
  float* xbuf = sim + E;                   // N*DIM
  float* xnew = xbuf + (long)N * DIM;      // N*DIM
  const long nd = (long)N * DIM;

  k_zero <<<1024, 256, 0, stream>>>(deg, (long)N);
  k_qnorm<<<1, DIM, 0, stream>>>(q, qn);
  k_deg  <<<(E + 255) / 256, 256, 0, stream>>>(dst, deg, E);

  {
    int waves  = (E + 15) / 16;
    int blocks = (waves + 7) / 8;        // 8 waves (256 thr) per block
    k_edge_sim<<<blocks, 256, 0, stream>>>(edge_attr, qn, sim, E);
  }

  const float* xc = x;
  for (int layer = 0; layer < 2; ++layer) {
    k_zero<<<4096, 256, 0, stream>>>(xnew, nd);
    long threads = (long)E * 32;
    k_scatter<<<(int)((threads + 255) / 256), 256, 0, stream>>>(xc, src, dst, sim, xnew, E);
    float* out = (layer == 0) ? xbuf : (float*)d_out;
    k_update<<<(int)((nd + 255) / 256), 256, 0, stream>>>(xc, xnew, deg, out, nd);
    xc = xbuf;
  }
}